// MixtureAlignmentLogLikelihood_15281493639185
// MI455X (gfx1250) — compile-verified
//
#include <hip/hip_runtime.h>
#include <math.h>
#include <stdint.h>

// Problem constants (match reference)
constexpr int LMAX = 5, TMAX = 50, KS = 65, KT = 49;
constexpr int TPAD = 64;   // t rows padded to 4 WMMA row-tiles
constexpr int KPAD = 52;   // KT padded to multiple of 4 (f32 WMMA K=4)
constexpr int TSTR = 52;   // LDS row stride for trg softmax
constexpr int SSTR = 66;   // LDS row stride for src softmax
#define EPSV 1e-30f

typedef float v2f __attribute__((ext_vector_type(2)));
typedef float v8f __attribute__((ext_vector_type(8)));
typedef int   v2i __attribute__((vector_size(8)));   // matches builtin param type

#define AS1 __attribute__((address_space(1)))
#define AS3 __attribute__((address_space(3)))

#if __has_builtin(__builtin_amdgcn_global_load_async_to_lds_b64) && \
    __has_builtin(__builtin_amdgcn_global_load_async_to_lds_b32)
#define ASYNC_BUILTINS 1
#else
#define ASYNC_BUILTINS 0
#endif

// Async global -> LDS copies (ASYNCcnt path). 8 bytes / 4 bytes per lane.
__device__ __forceinline__ void async_copy_b64(void* lds, const void* g) {
#if ASYNC_BUILTINS
    __builtin_amdgcn_global_load_async_to_lds_b64(
        (AS1 v2i*)(uintptr_t)g, (AS3 v2i*)(uintptr_t)lds, 0, 0);
#else
    asm volatile("global_load_async_to_lds_b64 %0, %1, off"
                 :: "v"((uint32_t)(uintptr_t)lds), "v"((uint64_t)(uintptr_t)g)
                 : "memory");
#endif
}
__device__ __forceinline__ void async_copy_b32(void* lds, const void* g) {
#if ASYNC_BUILTINS
    __builtin_amdgcn_global_load_async_to_lds_b32(
        (AS1 int*)(uintptr_t)g, (AS3 int*)(uintptr_t)lds, 0, 0);
#else
    asm volatile("global_load_async_to_lds_b32 %0, %1, off"
                 :: "v"((uint32_t)(uintptr_t)lds), "v"((uint64_t)(uintptr_t)g)
                 : "memory");
#endif
}
__device__ __forceinline__ void wait_async0() {
#if __has_builtin(__builtin_amdgcn_s_wait_asynccnt)
    __builtin_amdgcn_s_wait_asynccnt(0);
#else
    asm volatile("s_wait_asynccnt 0x0" ::: "memory");
#endif
}

__global__ __launch_bounds__(128) void maf_fused_kernel(
    const float* __restrict__ src_sent,   // [B, L, KS]
    const float* __restrict__ trg_sent,   // [B, T, KT]
    const float* __restrict__ src_bound,  // [B, L]
    const float* __restrict__ trg_bound,  // [B, T]
    float* __restrict__ scales_out,       // [B, T]  (d_out + 1)
    float* __restrict__ ll_ws)            // [B] partial log-likelihoods
{
    __shared__ float trg_raw[TMAX * KT];    // async staging (2450 floats)
    __shared__ float src_raw[LMAX * KS];    // async staging (325 floats)
    __shared__ float trg_soft[TPAD][TSTR];  // softmax(trg), zero padded
    __shared__ float src_soft[LMAX][SSTR];  // softmax(src)
    __shared__ float px[TPAD];              // (trg_soft @ P_st^T) column (all cols equal)
    __shared__ float scbuf[TMAX];           // per-step scales

    const int b    = blockIdx.x;
    const int tid  = threadIdx.x;
    const int lane = tid & 31;
    const int wave = tid >> 5;

    // ---------------- Phase A0: async stage this batch's rows into LDS ----------------
    // trg block: 2450 floats, batch offset 9800 B (8B aligned every b) -> b64 chunks.
    {
        const float* tg = trg_sent + (size_t)b * (TMAX * KT);
        for (int i = tid; i < (TMAX * KT) / 2; i += 128)       // 1225 chunks
            async_copy_b64(&trg_raw[2 * i], tg + 2 * i);
        const float* sg = src_sent + (size_t)b * (LMAX * KS);  // 1300 B: 4B aligned only
        for (int i = tid; i < LMAX * KS; i += 128)             // 325 dwords
            async_copy_b32(&src_raw[i], sg + i);
    }
    wait_async0();
    __syncthreads();

    // ---------------- Phase A: softmaxes (LDS -> LDS) ----------------
    if (tid < TPAD) {
        float* row = &trg_soft[tid][0];
        if (tid < TMAX) {
            const float* g = &trg_raw[tid * KT];
            float m = -3.402823466e38f;
            for (int q = 0; q < KT; ++q) m = fmaxf(m, g[q]);
            float s = 0.f;
            for (int q = 0; q < KT; ++q) { float e = expf(g[q] - m); row[q] = e; s += e; }
            float inv = 1.0f / s;
            for (int q = 0; q < KT; ++q) row[q] *= inv;
            for (int q = KT; q < KPAD; ++q) row[q] = 0.f;   // K padding
        } else {
            for (int q = 0; q < KPAD; ++q) row[q] = 0.f;    // row padding
        }
    } else if (tid < TPAD + LMAX) {
        const int l = tid - TPAD;
        const float* g = &src_raw[l * KS];
        float m = -3.402823466e38f;
        for (int q = 0; q < KS; ++q) m = fmaxf(m, g[q]);
        float s = 0.f;
        for (int q = 0; q < KS; ++q) { float e = expf(g[q] - m); src_soft[l][q] = e; s += e; }
        float inv = 1.0f / s;
        for (int q = 0; q < KS; ++q) src_soft[l][q] *= inv;
    }
    __syncthreads();

    // ---------------- Phase B: px via V_WMMA_F32_16X16X4_F32 ----------------
    // px[t, s] = sum_q trg_soft[t,q] * P_st[s,q], P_st == 1/KT (reference constant).
    // Each wave owns a 16-row tile; K-loop of 13 over KPAD=52 (A zero-padded).
    {
        const int t0 = wave * 16;
        const int m  = lane & 15;   // row within tile
        const int kk = lane >> 4;   // K half: lanes 0-15 -> K={0,1}, 16-31 -> K={2,3}
        v8f c = {0.f, 0.f, 0.f, 0.f, 0.f, 0.f, 0.f, 0.f};
        v2f bm; bm[0] = 1.0f / (float)KT; bm[1] = 1.0f / (float)KT; // uniform P_st^T tile
        for (int k0 = 0; k0 < KPAD; k0 += 4) {
            v2f a;
            a[0] = trg_soft[t0 + m][k0 + 2 * kk];
            a[1] = trg_soft[t0 + m][k0 + 2 * kk + 1];
            c = __builtin_amdgcn_wmma_f32_16x16x4_f32(
                    false, a, false, bm, (short)0, c, false, false);
        }
        // D layout: VGPR r -> (lanes 0-15: M=r, N=lane) (lanes 16-31: M=8+r, N=lane-16).
        // All N columns identical (uniform B); take N=0 (lanes 0 and 16).
        if (m == 0) {
            const int rbase = t0 + kk * 8;
            #pragma unroll
            for (int r = 0; r < 8; ++r) px[rbase + r] = c[r];
        }
    }
    __syncthreads();

    // ---------------- Phase C: ll partial + serial HMM recursion (wave 0) ----------------
    if (wave == 0) {
        float mask[LMAX], msum = 0.f;
        #pragma unroll
        for (int l = 0; l < LMAX; ++l) { mask[l] = src_bound[(size_t)b * LMAX + l]; msum += mask[l]; }
        const float Linv = 1.0f / fmaxf(msum, 1.0f);
        float diag[LMAX];
        #pragma unroll
        for (int l = 0; l < LMAX; ++l) diag[l] = mask[l] * Linv;

        // pzsum = sum_s sum_l mask_l * src_soft[l,s] / msum   (prob_z total mass)
        float p = 0.f;
        #pragma unroll
        for (int j = 0; j < 3; ++j) {
            int s = lane + 32 * j;
            if (s < KS) {
                float col = 0.f;
                #pragma unroll
                for (int l = 0; l < LMAX; ++l) col += mask[l] * src_soft[l][s];
                p += col;
            }
        }
        #pragma unroll
        for (int off = 16; off > 0; off >>= 1) p += __shfl_xor(p, off, 32);
        const float pzsum = p / msum;

        // ll_b = sum_t tb * log(max(pzsum * px_t, EPS))
        float ll = 0.f;
        for (int t = lane; t < TMAX; t += 32)
            ll += trg_bound[(size_t)b * TMAX + t] * logf(fmaxf(pzsum * px[t], EPSV));
        #pragma unroll
        for (int off = 16; off > 0; off >>= 1) ll += __shfl_xor(ll, off, 32);
        if (lane == 0) ll_ws[b] = ll;

        // Forward recursion. State f[l][s], s = lane + 32*j (j<3, valid s<KS).
        float f[LMAX][3], sv[LMAX][3];
        bool valid[3];
        #pragma unroll
        for (int j = 0; j < 3; ++j) valid[j] = (lane + 32 * j) < KS;
        #pragma unroll
        for (int l = 0; l < LMAX; ++l)
            #pragma unroll
            for (int j = 0; j < 3; ++j)
                sv[l][j] = valid[j] ? src_soft[l][lane + 32 * j] : 0.f;

        // t = 0: f0 = diag * src * px0
        {
            const float px0 = px[0];
            float acc = 0.f;
            #pragma unroll
            for (int l = 0; l < LMAX; ++l)
                #pragma unroll
                for (int j = 0; j < 3; ++j) { float v = diag[l] * sv[l][j] * px0; f[l][j] = v; acc += v; }
            #pragma unroll
            for (int off = 16; off > 0; off >>= 1) acc += __shfl_xor(acc, off, 32);
            if (lane == 0) scbuf[0] = acc;
            const float inv = 1.0f / fmaxf(acc, EPSV);
            #pragma unroll
            for (int l = 0; l < LMAX; ++l)
                #pragma unroll
                for (int j = 0; j < 3; ++j) f[l][j] *= inv;
        }
        for (int t = 1; t < TMAX; ++t) {
            const float pt = px[t];
            float fsum[LMAX];
            #pragma unroll
            for (int l = 0; l < LMAX; ++l) {
                float s = f[l][0] + f[l][1] + f[l][2];
                #pragma unroll
                for (int off = 16; off > 0; off >>= 1) s += __shfl_xor(s, off, 32);
                fsum[l] = s;
            }
            float S = 0.f;
            #pragma unroll
            for (int l = 0; l < LMAX; ++l) S += mask[l] * fsum[l];
            float vcoef[LMAX];
            #pragma unroll
            for (int l = 0; l < LMAX; ++l) vcoef[l] = Linv * mask[l] * S - diag[l] * fsum[l];

            float acc = 0.f;
            #pragma unroll
            for (int l = 0; l < LMAX; ++l)
                #pragma unroll
                for (int j = 0; j < 3; ++j) {
                    float fn = pt * (diag[l] * f[l][j] + vcoef[l] * sv[l][j]);
                    f[l][j] = fn; acc += fn;
                }
            #pragma unroll
            for (int off = 16; off > 0; off >>= 1) acc += __shfl_xor(acc, off, 32);
            if (lane == 0) scbuf[t] = acc;
            const float inv = 1.0f / fmaxf(acc, EPSV);
            #pragma unroll
            for (int l = 0; l < LMAX; ++l)
                #pragma unroll
                for (int j = 0; j < 3; ++j) f[l][j] *= inv;
        }
    }
    __syncthreads();

    if (tid < TMAX)
        scales_out[(size_t)b * TMAX + tid] = scbuf[tid];
}

// Deterministic fixed-order tree reduction of per-batch ll partials.
__global__ __launch_bounds__(256) void maf_ll_reduce(const float* __restrict__ ws,
                                                     float* __restrict__ out, int n)
{
    __shared__ float red[256];
    float s = 0.f;
    for (int i = threadIdx.x; i < n; i += 256) s += ws[i];
    red[threadIdx.x] = s;
    __syncthreads();
    for (int st = 128; st > 0; st >>= 1) {
        if ((int)threadIdx.x < st) red[threadIdx.x] += red[threadIdx.x + st];
        __syncthreads();
    }
    if (threadIdx.x == 0) out[0] = red[0];
}

extern "C" void kernel_launch(void* const* d_in, const int* in_sizes, int n_in,
                              void* d_out, int out_size, void* d_ws, size_t ws_size,
                              hipStream_t stream) {
    const float* src_sent  = (const float*)d_in[0];
    const float* trg_sent  = (const float*)d_in[1];
    const float* src_bound = (const float*)d_in[2];
    const float* trg_bound = (const float*)d_in[3];
    float* out = (float*)d_out;            // out[0] = ll, out[1..] = scales [B,T]
    float* ws  = (float*)d_ws;             // [B] ll partials

    const int B = in_sizes[0] / (LMAX * KS);

    maf_fused_kernel<<<B, 128, 0, stream>>>(src_sent, trg_sent, src_bound, trg_bound,
                                            out + 1, ws);
    maf_ll_reduce<<<1, 256, 0, stream>>>(ws, out, B);
}